// GPSAttentionLayer_65755949302335
// MI455X (gfx1250) — compile-verified
//
#include <hip/hip_runtime.h>
#include <math.h>

typedef float v2f __attribute__((ext_vector_type(2)));
typedef float v8f __attribute__((ext_vector_type(8)));

#define F_DIM 256
#define O_DIM 256
#define A_DIM 16
#define D_DIM 16
#define LDS_STRIDE 258   // 256 + 2 pad: conflict-free ds_load_b64 fragment reads
#define BN_BLOCKS 128

// ---------------------------------------------------------------------------
// FP32 GEMM via V_WMMA_F32_16X16X4_F32.  C[M,O] = A[M,K] * B[K,O]
// grid.x = M/16 row tiles, grid.y = O/16 col tiles, block = 1 wave (32).
// ---------------------------------------------------------------------------
__global__ __launch_bounds__(32)
void gps_wmma_gemm_f32(const float* __restrict__ A, const float* __restrict__ B,
                       float* __restrict__ C, int K, int Ocols) {
    __shared__ float ldsA[16 * LDS_STRIDE];
    const int lane = threadIdx.x;
    const int row0 = blockIdx.x * 16;
    const int n0   = blockIdx.y * 16;

    // Stage the 16 x K A-tile into LDS (float4 cooperative copy).
    const int f4PerRow = K >> 2;
    for (int i = lane; i < 16 * f4PerRow; i += 32) {
        const int r  = i / f4PerRow;
        const int c4 = i - r * f4PerRow;
        const float4 v = *(const float4*)(A + (size_t)(row0 + r) * K + (size_t)c4 * 4);
        float* dst = &ldsA[r * LDS_STRIDE + c4 * 4];
        dst[0] = v.x; dst[1] = v.y; dst[2] = v.z; dst[3] = v.w;
    }
    __syncthreads();

    v8f acc = {0.f, 0.f, 0.f, 0.f, 0.f, 0.f, 0.f, 0.f};
    const int m    = lane & 15;
    const int koff = (lane >> 4) << 1;           // 0 for lanes 0-15, 2 for 16-31
    const float* arow = &ldsA[m * LDS_STRIDE + koff];
    const float* bcol = B + n0 + m;

    #pragma unroll 8
    for (int k = 0; k < K; k += 4) {
        // A frag: lanes 0-15 hold K = k,k+1 ; lanes 16-31 hold K = k+2,k+3
        const v2f af = *(const v2f*)(arow + k);
        // B frag: same K split; N = lane&15
        const int kb = k + koff;
        v2f bf;
        bf[0] = bcol[(size_t)kb * Ocols];
        bf[1] = bcol[(size_t)(kb + 1) * Ocols];
        acc = __builtin_amdgcn_wmma_f32_16x16x4_f32(
            false, af, false, bf, (short)0, acc, false, false);
    }

    // C layout: VGPR r -> M = r (lanes 0-15) / r+8 (lanes 16-31); N = lane&15
    float* crow = C + (size_t)(row0 + (lane >> 4) * 8) * Ocols + n0 + m;
    #pragma unroll
    for (int r = 0; r < 8; ++r)
        crow[(size_t)r * Ocols] = acc[r];
}

// ---------------------------------------------------------------------------
// Per-node softmax attention + weighted neighbor aggregation (one wave/node).
// fh[n] = (accumulate ? fh[n] : base[n]) + sum_d w[n,d] * newh[rf[n,d]]
// ---------------------------------------------------------------------------
__global__ __launch_bounds__(256)
void gps_attn_aggregate(const float* __restrict__ Key, const float* __restrict__ Query,
                        const int* __restrict__ rf, const float* __restrict__ newh,
                        const float* __restrict__ base, float* __restrict__ fh,
                        int N, int accumulate) {
    const int lane = threadIdx.x & 31;
    const int node = blockIdx.x * 8 + (threadIdx.x >> 5);
    if (node >= N) return;

    float att = -1e30f;
    int   idx = 0;
    if (lane < D_DIM) {
        idx = rf[(size_t)node * D_DIM + lane];
        const float4* kp = (const float4*)(Key + (size_t)node * A_DIM);
        const float4* qp = (const float4*)(Query + (size_t)idx * A_DIM);
        float s = 0.f;
        #pragma unroll
        for (int j = 0; j < 4; ++j) {
            const float4 a = kp[j], b = qp[j];
            s += a.x * b.x + a.y * b.y + a.z * b.z + a.w * b.w;
        }
        att = (idx != N - 1) ? s : -1e30f;   // masked -> softmax weight 0
    }
    // row max (all-masked row -> att-mx == 0 -> uniform weights, as reference)
    float mx = att;
    #pragma unroll
    for (int off = 16; off > 0; off >>= 1)
        mx = fmaxf(mx, __shfl_xor(mx, off, 32));
    float e = (lane < D_DIM) ? expf(att - mx) : 0.f;
    float sum = e;
    #pragma unroll
    for (int off = 16; off > 0; off >>= 1)
        sum += __shfl_xor(sum, off, 32);
    const float w = e / sum;

    // Aggregate 256 columns: 8 per lane, coalesced float4 gathers.
    const int c0 = lane * 8;
    const float* src = (accumulate ? fh : base) + (size_t)node * O_DIM + c0;
    float4 a0 = *(const float4*)(src);
    float4 a1 = *(const float4*)(src + 4);
    #pragma unroll
    for (int d = 0; d < D_DIM; ++d) {
        const float wd = __shfl(w, d, 32);
        const int   id = __shfl(idx, d, 32);
        const float4* rp = (const float4*)(newh + (size_t)id * O_DIM + c0);
        const float4 v0 = rp[0], v1 = rp[1];
        a0.x += wd * v0.x; a0.y += wd * v0.y; a0.z += wd * v0.z; a0.w += wd * v0.w;
        a1.x += wd * v1.x; a1.y += wd * v1.y; a1.z += wd * v1.z; a1.w += wd * v1.w;
    }
    float* dst = fh + (size_t)node * O_DIM + c0;
    *(float4*)(dst)     = a0;
    *(float4*)(dst + 4) = a1;
}

// ---------------------------------------------------------------------------
// BatchNorm: deterministic two-stage column reduction + in-place apply.
// ---------------------------------------------------------------------------
__global__ __launch_bounds__(256)
void gps_bn_partial(const float* __restrict__ fh, float* __restrict__ psum,
                    float* __restrict__ psumsq, int N) {
    const int c = threadIdx.x;
    float s = 0.f, sq = 0.f;
    for (int r = blockIdx.x; r < N; r += gridDim.x) {
        const float v = fh[(size_t)r * O_DIM + c];
        s += v; sq += v * v;
    }
    psum[blockIdx.x * O_DIM + c]   = s;
    psumsq[blockIdx.x * O_DIM + c] = sq;
}

__global__ __launch_bounds__(256)
void gps_bn_finalize(const float* __restrict__ psum, const float* __restrict__ psumsq,
                     float* __restrict__ stats, int N, int nblk) {
    const int c = threadIdx.x;
    float s = 0.f, sq = 0.f;
    for (int b = 0; b < nblk; ++b) {
        s  += psum[b * O_DIM + c];
        sq += psumsq[b * O_DIM + c];
    }
    const float mu  = s / (float)N;
    const float var = sq / (float)N - mu * mu;   // biased, as jnp.var
    stats[c]         = mu;
    stats[O_DIM + c] = rsqrtf(var + 1e-5f);
}

__global__ __launch_bounds__(256)
void gps_bn_apply(float* __restrict__ fh, const float* __restrict__ stats,
                  const float* __restrict__ gamma, const float* __restrict__ beta,
                  int total) {
    const int stride = gridDim.x * blockDim.x;
    for (int i = blockIdx.x * blockDim.x + threadIdx.x; i < total; i += stride) {
        const int c = i & (O_DIM - 1);
        fh[i] = (fh[i] - stats[c]) * stats[O_DIM + c] * gamma[c] + beta[c];
    }
}

// ---------------------------------------------------------------------------
// Copy existing receptive-field planes (int -> float) into the output tail.
// ---------------------------------------------------------------------------
__global__ void gps_copy_rf(const int* __restrict__ rf, float* __restrict__ out, int n) {
    const int i = blockIdx.x * blockDim.x + threadIdx.x;
    if (i < n) out[i] = (float)rf[i];
}

// ---------------------------------------------------------------------------
// Receptive-field expansion: 256 candidates/node, top-16 by attention with
// jax.lax.top_k tie semantics (lower candidate index wins). One wave/node.
// ---------------------------------------------------------------------------
__global__ __launch_bounds__(256)
void gps_expand_topk(const float* __restrict__ Key, const float* __restrict__ Query,
                     const int* __restrict__ rf_last, const int* __restrict__ adj,
                     float* __restrict__ out, int N) {
    const int lane = threadIdx.x & 31;
    const int node = blockIdx.x * 8 + (threadIdx.x >> 5);
    if (node >= N) return;

    const float4* kp = (const float4*)(Key + (size_t)node * A_DIM);
    const float4 k0 = kp[0], k1 = kp[1], k2 = kp[2], k3 = kp[3];

    float attv[8];
    int   nid[8];
    #pragma unroll
    for (int j = 0; j < 8; ++j) {
        const int m  = lane + 32 * j;                      // candidate index in [0,256)
        const int r  = rf_last[(size_t)node * D_DIM + (m >> 4)];
        const int nb = adj[(size_t)r * D_DIM + (m & 15)];
        const float4* qp = (const float4*)(Query + (size_t)nb * A_DIM);
        const float4 q0 = qp[0], q1 = qp[1], q2 = qp[2], q3 = qp[3];
        float s = k0.x * q0.x + k0.y * q0.y + k0.z * q0.z + k0.w * q0.w
                + k1.x * q1.x + k1.y * q1.y + k1.z * q1.z + k1.w * q1.w
                + k2.x * q2.x + k2.y * q2.y + k2.z * q2.z + k2.w * q2.w
                + k3.x * q3.x + k3.y * q3.y + k3.z * q3.z + k3.w * q3.w;
        attv[j] = (nb != N - 1) ? s : -1e30f;  // masked below all real scores
        nid[j]  = nb;
    }

    for (int t = 0; t < D_DIM; ++t) {
        // local best (smallest j on tie -> smallest m for this lane)
        float bv = -3.0e38f; int bj = 0; int bn = 0;
        #pragma unroll
        for (int j = 0; j < 8; ++j)
            if (attv[j] > bv) { bv = attv[j]; bj = j; bn = nid[j]; }
        float v = bv; int mm = lane + 32 * bj; int nn = bn;
        // wave argmax, tie -> smaller candidate index (jax top_k order)
        #pragma unroll
        for (int off = 16; off > 0; off >>= 1) {
            const float ov = __shfl_xor(v, off, 32);
            const int   om = __shfl_xor(mm, off, 32);
            const int   on = __shfl_xor(nn, off, 32);
            if (ov > v || (ov == v && om < mm)) { v = ov; mm = om; nn = on; }
        }
        if (lane == 0) out[(size_t)node * D_DIM + t] = (float)nn;
        // retire winner
        const int wl = mm & 31, wj = mm >> 5;
        if (lane == wl) {
            #pragma unroll
            for (int j = 0; j < 8; ++j)
                if (j == wj) attv[j] = -3.0e38f;
        }
    }
}

// ---------------------------------------------------------------------------
extern "C" void kernel_launch(void* const* d_in, const int* in_sizes, int n_in,
                              void* d_out, int out_size, void* d_ws, size_t ws_size,
                              hipStream_t stream) {
    const float* input = (const float*)d_in[0];        // [N,256]
    const int*   rf    = (const int*)d_in[1];          // [2,N,16]
    const int*   adj   = (const int*)d_in[2];          // [N,16]
    const float* W     = (const float*)d_in[3];        // [2,256,256]
    const float* Wk    = (const float*)d_in[4];        // [256,16]
    const float* Wq    = (const float*)d_in[5];        // [256,16]
    const float* gamma = (const float*)d_in[6];        // [256]
    const float* beta  = (const float*)d_in[7];        // [256]

    const int N = in_sizes[0] / F_DIM;                 // 30000

    // workspace layout (floats)
    float* ws     = (float*)d_ws;
    float* newh0  = ws;                                // N*256
    float* newh1  = newh0 + (size_t)N * O_DIM;         // N*256
    float* KeyB   = newh1 + (size_t)N * O_DIM;         // N*16
    float* QueryB = KeyB + (size_t)N * A_DIM;          // N*16
    float* psum   = QueryB + (size_t)N * A_DIM;        // BN_BLOCKS*256
    float* psumsq = psum + (size_t)BN_BLOCKS * O_DIM;  // BN_BLOCKS*256
    float* stats  = psumsq + (size_t)BN_BLOCKS * O_DIM;// 512

    // output layout (floats)
    float* fh     = (float*)d_out;                     // [N,256] final_h
    float* out_rf = fh + (size_t)N * O_DIM;            // [3,N,16]

    const int mt = N / 16;                             // row tiles (N % 16 == 0 here)

    // 1) WMMA GEMMs: new_h[0], new_h[1], Key, Query
    gps_wmma_gemm_f32<<<dim3(mt, O_DIM / 16), 32, 0, stream>>>(input, W,                 newh0, F_DIM, O_DIM);
    gps_wmma_gemm_f32<<<dim3(mt, O_DIM / 16), 32, 0, stream>>>(input, W + F_DIM * O_DIM, newh1, F_DIM, O_DIM);
    gps_wmma_gemm_f32<<<dim3(mt, 1),          32, 0, stream>>>(input, Wk, KeyB,   F_DIM, A_DIM);
    gps_wmma_gemm_f32<<<dim3(mt, 1),          32, 0, stream>>>(input, Wq, QueryB, F_DIM, A_DIM);

    // 2) attention + aggregation per thickness (fh = newh0 + c0 + c1)
    const int nodeBlocks = (N + 7) / 8;
    gps_attn_aggregate<<<nodeBlocks, 256, 0, stream>>>(KeyB, QueryB, rf,
                                                       newh0, newh0, fh, N, 0);
    gps_attn_aggregate<<<nodeBlocks, 256, 0, stream>>>(KeyB, QueryB, rf + (size_t)N * D_DIM,
                                                       newh1, newh0, fh, N, 1);

    // 3) BatchNorm (training-mode batch stats), in place on fh
    gps_bn_partial<<<BN_BLOCKS, 256, 0, stream>>>(fh, psum, psumsq, N);
    gps_bn_finalize<<<1, 256, 0, stream>>>(psum, psumsq, stats, N, BN_BLOCKS);
    gps_bn_apply<<<480, 256, 0, stream>>>(fh, stats, gamma, beta, N * O_DIM);

    // 4) receptive_field_out = [rf[0], rf[1], expand]
    const int rfElems = 2 * N * D_DIM;
    gps_copy_rf<<<(rfElems + 255) / 256, 256, 0, stream>>>(rf, out_rf, rfElems);
    gps_expand_topk<<<nodeBlocks, 256, 0, stream>>>(KeyB, QueryB, rf + (size_t)N * D_DIM,
                                                    adj, out_rf + (size_t)rfElems, N);
}